// ADMM_Net_59820304499474
// MI455X (gfx1250) — compile-verified
//
#include <hip/hip_runtime.h>
#include <hip/hip_bf16.h>
#include <math.h>

// ============================================================================
// ADMM deconvolution (DiffuserCam) for MI455X / gfx1250, wave32.
//
// All 2-D FFTs are 1024x1024, done as row-FFT passes with transposed writes.
// 1024-point line FFT = radix-4 (VALU) + two radix-16 stages implemented as
// 16x16 complex matrix products on the WMMA unit (v_wmma_f32_16x16x4_f32,
// 4 chained K=4 WMMAs per real 16x16x16 matmul, 16 WMMAs per complex tile).
//
// Workspace requirement: ~587 MB (17 full planes + 4 x 4 MB of filters).
// ============================================================================

#define DIM      1024
#define BATCH    8
#define NPLANE   (BATCH * DIM * DIM)          // 8,388,608 floats per full field
#define MPLANE   (DIM * DIM)                  // 1,048,576 (single 2D plane)
#define NT       128                          // threads per FFT block (4 waves)
#define PI2      6.28318530717958647692f

#define MU1f  1.0e-6f
#define MU2f  1.0e-5f
#define MU3f  1.0e-5f
#define TAUf  0.1f

typedef float v2f __attribute__((ext_vector_type(2)));
typedef float v8f __attribute__((ext_vector_type(8)));
typedef int   v4i __attribute__((vector_size(16)));

#if __has_builtin(__builtin_amdgcn_wmma_f32_16x16x4_f32)
#define HAVE_WMMA_F32 1
#endif

#if __has_builtin(__builtin_amdgcn_global_load_async_to_lds_b128) && \
    __has_builtin(__builtin_amdgcn_s_wait_asynccnt)
#define HAVE_ASYNC_LDS 1
#endif

// ---------------------------------------------------------------------------
// Async (or plain) load of one 1024-float line from global into LDS.
// Caller must __syncthreads() afterwards.
// ---------------------------------------------------------------------------
__device__ __forceinline__ void lds_line_load(float* dst, const float* __restrict__ src)
{
    const int tid = threadIdx.x;
#if defined(HAVE_ASYNC_LDS)
#pragma unroll
    for (int rep = 0; rep < 2; ++rep) {
        const int i16 = tid + rep * NT;   // 0..255 x 16B = 4 KB line
        __builtin_amdgcn_global_load_async_to_lds_b128(
            (__attribute__((address_space(1))) v4i*)((float*)src + 4 * i16),
            (__attribute__((address_space(3))) v4i*)(dst + 4 * i16),
            0, 0);
    }
    __builtin_amdgcn_s_wait_asynccnt(0);
#else
    for (int i = tid; i < DIM; i += NT) dst[i] = src[i];
#endif
}

// ---------------------------------------------------------------------------
// One complex 16x16 DFT applied to a 16-column tile held in LDS (row-major,
// row stride 64). Each wave handles one tile (colBase = waveId*16).
//
//  C(16x16) = W16(16x16) * D(16x16),  W16[m][k] = exp(sign*2*pi*i*m*k/16)
//
// WMMA operand layouts (CDNA5 ISA 7.12.2, 32-bit, wave32):
//  A 16x4 : lane L: M=L&15 ; VGPR v holds K = 4*k0 + 2*(L>>4) + v
//  B 4x16 : lane L: N=L&15 ; VGPR v holds K = 4*k0 + 2*(L>>4) + v
//  C/D    : VGPR j: M = j + 8*(L>>4), N = L&15
//
// dstMode 0 (middle stage): out index = n1*64 + (kv + 4*row), col=(n1*4+kv),
//                           optional W1024^{n1*k2} twiddle.
// dstMode 1 (final stage) : out index = col + 64*row (digit-ordered line).
// ---------------------------------------------------------------------------
static __device__ __forceinline__ void dft16_tile(
    const float* __restrict__ sr, const float* __restrict__ si,
    float* __restrict__ dr, float* __restrict__ di,
    int colBase, float sign, int dstMode, float scale, int tw1024)
{
    const int lane = threadIdx.x & 31;
    const int hi   = lane >> 4;
    const int nloc = lane & 15;
    const int col  = colBase + nloc;

    v8f cr = {0.f,0.f,0.f,0.f,0.f,0.f,0.f,0.f};
    v8f ci = {0.f,0.f,0.f,0.f,0.f,0.f,0.f,0.f};

#if defined(HAVE_WMMA_F32)
#pragma unroll
    for (int k0 = 0; k0 < 4; ++k0) {
        const int kA = 4 * k0 + 2 * hi;
        float s0, c0, s1, c1;
        __sincosf(sign * PI2 * (float)(nloc * (kA + 0)) * (1.0f / 16.0f), &s0, &c0);
        __sincosf(sign * PI2 * (float)(nloc * (kA + 1)) * (1.0f / 16.0f), &s1, &c1);
        v2f ar  = { c0,  c1 };
        v2f ai  = { s0,  s1 };
        v2f ain = { -s0, -s1 };            // f32 WMMA has no A-negate: fold it here
        v2f br  = { sr[(kA + 0) * 64 + col], sr[(kA + 1) * 64 + col] };
        v2f bi  = { si[(kA + 0) * 64 + col], si[(kA + 1) * 64 + col] };

        // Re += Ar*Br - Ai*Bi ; Im += Ar*Bi + Ai*Br
        cr = __builtin_amdgcn_wmma_f32_16x16x4_f32(false, ar,  false, br, (short)0, cr, false, false);
        cr = __builtin_amdgcn_wmma_f32_16x16x4_f32(false, ain, false, bi, (short)0, cr, false, false);
        ci = __builtin_amdgcn_wmma_f32_16x16x4_f32(false, ar,  false, bi, (short)0, ci, false, false);
        ci = __builtin_amdgcn_wmma_f32_16x16x4_f32(false, ai,  false, br, (short)0, ci, false, false);
    }
#else
    // Scalar fallback (no WMMA builtin): direct 16-term sums per C element.
#pragma unroll
    for (int j = 0; j < 8; ++j) {
        const int m = j + 8 * hi;
        float accR = 0.f, accI = 0.f;
        for (int k = 0; k < 16; ++k) {
            float s, c;
            __sincosf(sign * PI2 * (float)(m * k) * (1.0f / 16.0f), &s, &c);
            const float br = sr[k * 64 + col], bi = si[k * 64 + col];
            accR += c * br - s * bi;
            accI += c * bi + s * br;
        }
        cr[j] = accR; ci[j] = accI;
    }
#endif

#pragma unroll
    for (int j = 0; j < 8; ++j) {
        const int row = j + 8 * hi;
        float re = cr[j], im = ci[j];
        int dstIdx;
        if (dstMode == 0) {
            const int n1 = col >> 2;
            const int kv = col & 3;
            const int k2 = kv + 4 * row;
            if (tw1024) {                          // six-step twiddle W1024^{n1*k2}
                float s, c;
                __sincosf(sign * PI2 * (float)(n1 * k2) * (1.0f / 1024.0f), &s, &c);
                const float tr = re * c - im * s;
                im = re * s + im * c;
                re = tr;
            }
            dstIdx = n1 * 64 + k2;
        } else {
            dstIdx = col + 64 * row;               // final line: k = k2 + 64*k1
        }
        dr[dstIdx] = re * scale;
        di[dstIdx] = im * scale;
    }
}

// ---------------------------------------------------------------------------
// 1024-point complex FFT of one line held in LDS. 1024 = 4 x 16 x 16.
//   x[n], n = n1 + 16*u + 256*v   (n1,u in [0,16), v in [0,4))
//   step1: radix-4 over v, twiddle W64^{u*kv}        (VALU)
//   step2: DFT-16 over u, twiddle W1024^{n1*k2}      (WMMA)
//   step3: DFT-16 over n1 -> X[k2 + 64*k1]           (WMMA)
// Input in (sr,si); result in (dr,di); both buffers are clobbered.
// sign = -1 forward, +1 inverse; `scale` applied at the last stage.
// ---------------------------------------------------------------------------
static __device__ void fft1024_lds(float* sr, float* si, float* dr, float* di,
                                   float sign, float scale)
{
    const int tid = threadIdx.x;

    // ---- step 1: radix-4 + W64 twiddle : (sr,si) -> (dr,di) as T[u][n1*4+kv]
#pragma unroll
    for (int rep = 0; rep < 2; ++rep) {
        const int g  = tid + NT * rep;       // 256 (n1,u) groups
        const int n1 = g & 15;
        const int u  = g >> 4;
        const int b0 = n1 + 16 * u;
        const float t0r = sr[b0 +   0], t0i = si[b0 +   0];
        const float t1r = sr[b0 + 256], t1i = si[b0 + 256];
        const float t2r = sr[b0 + 512], t2i = si[b0 + 512];
        const float t3r = sr[b0 + 768], t3i = si[b0 + 768];
        const float e0r = t0r + t2r, e0i = t0i + t2i;
        const float e1r = t0r - t2r, e1i = t0i - t2i;
        const float o0r = t1r + t3r, o0i = t1i + t3i;
        const float o1r = t1r - t3r, o1i = t1i - t3i;
        float yr[4], yi[4];
        yr[0] = e0r + o0r;            yi[0] = e0i + o0i;
        yr[2] = e0r - o0r;            yi[2] = e0i - o0i;
        yr[1] = e1r - sign * o1i;     yi[1] = e1i + sign * o1r;   // e1 + sign*i*o1
        yr[3] = e1r + sign * o1i;     yi[3] = e1i - sign * o1r;
#pragma unroll
        for (int kv = 0; kv < 4; ++kv) {
            float s, c;
            __sincosf(sign * PI2 * (float)(u * kv) * (1.0f / 64.0f), &s, &c);
            dr[u * 64 + n1 * 4 + kv] = yr[kv] * c - yi[kv] * s;
            di[u * 64 + n1 * 4 + kv] = yr[kv] * s + yi[kv] * c;
        }
    }
    __syncthreads();

    const int wv = tid >> 5;                 // wave id 0..3, one 16-col tile each
    // ---- step 2: DFT-16 over u (WMMA), output S[n1][k2] into (sr,si)
    dft16_tile(dr, di, sr, si, wv * 16, sign, /*dstMode*/0, 1.0f, /*tw1024*/1);
    __syncthreads();
    // ---- step 3: DFT-16 over n1 (WMMA), final line into (dr,di)
    dft16_tile(sr, si, dr, di, wv * 16, sign, /*dstMode*/1, scale, 0);
    __syncthreads();
}

// ---------------------------------------------------------------------------
// Pass 1: forward row FFT of a real field, write complex transposed.
// ---------------------------------------------------------------------------
__global__ __launch_bounds__(NT) void k_fft_pass1(
    const float* __restrict__ in, float* __restrict__ oRe, float* __restrict__ oIm)
{
    __shared__ float lds[4 * DIM];
    float *ar = lds, *ai = lds + DIM, *br = lds + 2 * DIM, *bi = lds + 3 * DIM;
    const int img = blockIdx.x >> 10;
    const int row = blockIdx.x & 1023;
    const size_t base = ((size_t)img << 20) + ((size_t)row << 10);

    lds_line_load(ar, in + base);
    for (int i = threadIdx.x; i < DIM; i += NT) ai[i] = 0.0f;
    __syncthreads();

    fft1024_lds(ar, ai, br, bi, -1.0f, 1.0f);

    const size_t ob = ((size_t)img << 20) + (size_t)row;
    for (int i = threadIdx.x; i < DIM; i += NT) {
        oRe[ob + ((size_t)i << 10)] = br[i];
        oIm[ob + ((size_t)i << 10)] = bi[i];
    }
}

// ---------------------------------------------------------------------------
// Pass 2 (column dim): forward FFT, then either
//   storeSpec=1 : write spectrum * (-1)^(j+k) in line layout (builds Hf), or
//   multiply filter1 (optionally conjugated; f1Im==null => real filter),
//   inverse FFT, write transposed. If f2Re!=null, additionally produce a
//   second output filtered by f2 (complex), sharing the forward FFT.
// ---------------------------------------------------------------------------
__global__ __launch_bounds__(NT) void k_fft_pass2(
    const float* __restrict__ iRe, const float* __restrict__ iIm,
    float* __restrict__ oRe, float* __restrict__ oIm,
    const float* __restrict__ f1Re, const float* __restrict__ f1Im, int conj1,
    const float* __restrict__ f2Re, const float* __restrict__ f2Im,
    float* __restrict__ o2Re, float* __restrict__ o2Im,
    int storeSpec, float invScale)
{
    __shared__ float lds[6 * DIM];
    float *ar = lds,           *ai = lds + DIM;
    float *br = lds + 2 * DIM, *bi = lds + 3 * DIM;
    float *gr = lds + 4 * DIM, *gi = lds + 5 * DIM;
    const int img = blockIdx.x >> 10;
    const int j   = blockIdx.x & 1023;                 // k_c line index
    const size_t base = ((size_t)img << 20) + ((size_t)j << 10);

    lds_line_load(ar, iRe + base);
    lds_line_load(ai, iIm + base);
    __syncthreads();

    fft1024_lds(ar, ai, br, bi, -1.0f, 1.0f);          // spectrum line over k_r

    if (storeSpec) {                                   // Hf build: apply parity S
        for (int i = threadIdx.x; i < DIM; i += NT) {
            const float sgn = ((i + j) & 1) ? -1.0f : 1.0f;
            oRe[base + i] = sgn * br[i];
            oIm[base + i] = sgn * bi[i];
        }
        return;
    }

    const int fbase = j << 10;
    for (int i = threadIdx.x; i < DIM; i += NT) {
        const float xr = br[i], xi = bi[i];
        const float fr = f1Re[fbase + i];
        float       fi = f1Im ? f1Im[fbase + i] : 0.0f;
        if (conj1) fi = -fi;
        const float yr = xr * fr - xi * fi;
        const float yi = xr * fi + xi * fr;
        br[i] = yr; bi[i] = yi;
        if (f2Re) {                                    // second (chained) filter
            const float hr = f2Re[fbase + i], h2 = f2Im[fbase + i];
            gr[i] = yr * hr - yi * h2;
            gi[i] = yr * h2 + yi * hr;
        }
    }
    __syncthreads();

    fft1024_lds(br, bi, ar, ai, +1.0f, invScale);      // inverse -> (ar,ai)
    const size_t ob = ((size_t)img << 20) + (size_t)j;
    for (int i = threadIdx.x; i < DIM; i += NT) {
        oRe[ob + ((size_t)i << 10)] = ar[i];
        oIm[ob + ((size_t)i << 10)] = ai[i];
    }

    if (f2Re) {
        __syncthreads();
        fft1024_lds(gr, gi, br, bi, +1.0f, invScale);
        for (int i = threadIdx.x; i < DIM; i += NT) {
            o2Re[ob + ((size_t)i << 10)] = br[i];
            o2Im[ob + ((size_t)i << 10)] = bi[i];
        }
    }
}

// ---------------------------------------------------------------------------
// Pass 3: inverse row FFT, keep real part (completes real(ifft2)).
// ---------------------------------------------------------------------------
__global__ __launch_bounds__(NT) void k_fft_pass3(
    const float* __restrict__ iRe, const float* __restrict__ iIm,
    float* __restrict__ out, float invScale)
{
    __shared__ float lds[4 * DIM];
    float *ar = lds, *ai = lds + DIM, *br = lds + 2 * DIM, *bi = lds + 3 * DIM;
    const int img = blockIdx.x >> 10;
    const int row = blockIdx.x & 1023;
    const size_t base = ((size_t)img << 20) + ((size_t)row << 10);

    lds_line_load(ar, iRe + base);
    lds_line_load(ai, iIm + base);
    __syncthreads();

    fft1024_lds(ar, ai, br, bi, +1.0f, invScale);

    for (int i = threadIdx.x; i < DIM; i += NT) out[base + i] = br[i];
}

// ===========================================================================
// Element-wise kernels
// ===========================================================================
__device__ __forceinline__ float softf(float v, float t)
{
    return copysignf(fmaxf(fabsf(v) - t, 0.0f), v);
}

__global__ void k_embed_h(const float* __restrict__ h, float* __restrict__ hpad)
{
    const int idx = blockIdx.x * 256 + threadIdx.x;       // 512*512
    const int r = idx >> 9, c = idx & 511;
    hpad[((r + 256) << 10) + (c + 256)] = h[idx];
}

__global__ void k_make_rdiv(const float* __restrict__ hr, const float* __restrict__ hi,
                            float* __restrict__ rdv)
{
    const int idx = blockIdx.x * 256 + threadIdx.x;       // 1<<20
    const int k = idx & 1023, j = idx >> 10;
    const float mag = hr[idx] * hr[idx] + hi[idx] * hi[idx];
    const float ltl = 4.0f - 2.0f * __cosf(PI2 * (float)k * (1.0f / 1024.0f))
                           - 2.0f * __cosf(PI2 * (float)j * (1.0f / 1024.0f));
    rdv[idx] = 1.0f / (MU1f * mag + MU2f * fabsf(ltl) + MU3f);
}

// U = soft(psi(X) + a2/MU2, TAU/MU2)
__global__ void k_U(const float* __restrict__ X, const float* __restrict__ A2,
                    float* __restrict__ U)
{
    const int idx = blockIdx.x * 256 + threadIdx.x;
    const int b = idx >> 20, rc = idx & (MPLANE - 1);
    const int r = rc >> 10, c = rc & 1023;
    const int rm = (r == 0) ? 1023 : r - 1;
    const int cm = (c == 0) ? 1023 : c - 1;
    const float x  = X[idx];
    const float p0 = X[(b << 20) + (rm << 10) + c] - x;   // roll(+1, axis=-2) - x
    const float p1 = X[(b << 20) + (r << 10) + cm] - x;   // roll(+1, axis=-1) - x
    const float th = TAUf / MU2f;
    U[idx]          = softf(p0 + A2[idx]          * (1.0f / MU2f), th);
    U[NPLANE + idx] = softf(p1 + A2[NPLANE + idx] * (1.0f / MU2f), th);
}

// V = Vdiv * (a1 + MU1*H(X) + CT(y));   Vdiv, CT(y) computed analytically
__global__ void k_V(const float* __restrict__ A1, const float* __restrict__ HX,
                    const float* __restrict__ y, float* __restrict__ V)
{
    const int idx = blockIdx.x * 256 + threadIdx.x;
    const int b = idx >> 20, rc = idx & (MPLANE - 1);
    const int r = rc >> 10, c = rc & 1023;
    const bool inside = (r >= 256) & (r < 768) & (c >= 256) & (c < 768);
    const float cty  = inside ? y[((b << 9) + (r - 256)) * 512 + (c - 256)] : 0.0f;
    const float vdiv = inside ? 1.0f / (1.0f + MU1f) : 1.0f / MU1f;
    V[idx] = vdiv * (A1[idx] + MU1f * HX[idx] + cty);
}

__global__ void k_W(const float* __restrict__ A3, const float* __restrict__ X,
                    float* __restrict__ W)
{
    const int idx = blockIdx.x * 256 + threadIdx.x;
    W[idx] = fmaxf(A3[idx] * (1.0f / MU3f) + X[idx], 0.0f);
}

// q = MU1*V - a1   (input to Hadj pipeline)
__global__ void k_q(const float* __restrict__ V, const float* __restrict__ A1,
                    float* __restrict__ Q)
{
    const int idx = blockIdx.x * 256 + threadIdx.x;
    Q[idx] = MU1f * V[idx] - A1[idx];
}

// r += (MU3*W - a3) + psit(MU2*U - a2)   (r already holds Hadj(MU1*V - a1))
__global__ void k_r(const float* __restrict__ U, const float* __restrict__ A2,
                    const float* __restrict__ W, const float* __restrict__ A3,
                    float* __restrict__ R)
{
    const int idx = blockIdx.x * 256 + threadIdx.x;
    const int b = idx >> 20, rc = idx & (MPLANE - 1);
    const int r = rc >> 10, c = rc & 1023;
    const int rp = (r == 1023) ? 0 : r + 1;
    const int cp = (c == 1023) ? 0 : c + 1;
    const int iN0 = (b << 20) + (rp << 10) + c;
    const int iN1 = (b << 20) + (r << 10) + cp;
    const float q0  = MU2f * U[idx]          - A2[idx];
    const float q0n = MU2f * U[iN0]          - A2[iN0];
    const float q1  = MU2f * U[NPLANE + idx] - A2[NPLANE + idx];
    const float q1n = MU2f * U[NPLANE + iN1] - A2[NPLANE + iN1];
    R[idx] += (MU3f * W[idx] - A3[idx]) + (q0n - q0) + (q1n - q1);
}

// a1 += MU1*(H(Xnew) - V); a2 += MU2*(psi(Xnew) - U); a3 += MU3*(Xnew - W)
__global__ void k_a(const float* __restrict__ X, const float* __restrict__ U,
                    const float* __restrict__ V, const float* __restrict__ W,
                    const float* __restrict__ HX,
                    float* __restrict__ A1, float* __restrict__ A2,
                    float* __restrict__ A3)
{
    const int idx = blockIdx.x * 256 + threadIdx.x;
    const int b = idx >> 20, rc = idx & (MPLANE - 1);
    const int r = rc >> 10, c = rc & 1023;
    const int rm = (r == 0) ? 1023 : r - 1;
    const int cm = (c == 0) ? 1023 : c - 1;
    const float x  = X[idx];
    const float p0 = X[(b << 20) + (rm << 10) + c] - x;
    const float p1 = X[(b << 20) + (r << 10) + cm] - x;
    A1[idx]          += MU1f * (HX[idx] - V[idx]);
    A2[idx]          += MU2f * (p0 - U[idx]);
    A2[NPLANE + idx] += MU2f * (p1 - U[NPLANE + idx]);
    A3[idx]          += MU3f * (x - W[idx]);
}

__global__ void k_crop(const float* __restrict__ X, float* __restrict__ out)
{
    const int idx = blockIdx.x * 256 + threadIdx.x;       // 8*512*512
    const int b = idx >> 18, rc = idx & ((1 << 18) - 1);
    const int r = rc >> 9, c = rc & 511;
    out[idx] = X[(b << 20) + ((r + 256) << 10) + (c + 256)];
}

// ===========================================================================
// Host driver
// ===========================================================================
extern "C" void kernel_launch(void* const* d_in, const int* in_sizes, int n_in,
                              void* d_out, int out_size, void* d_ws, size_t ws_size,
                              hipStream_t stream)
{
    const float* y = (const float*)d_in[0];   // (8,1,512,512) f32
    const float* h = (const float*)d_in[1];   // (512,512)     f32
    float* out = (float*)d_out;               // (8,1,512,512) f32
    float* ws  = (float*)d_ws;

    const size_t N = (size_t)NPLANE;
    const size_t M = (size_t)MPLANE;

    float* X   = ws;            // full-field real state
    float* U   = X   + N;       // 2 planes
    float* V   = U   + 2 * N;
    float* W   = V   + N;
    float* A1  = W   + N;
    float* A2  = A1  + N;       // 2 planes
    float* A3  = A2  + 2 * N;
    float* HX  = A3  + N;       // cached H(X) (real)
    float* RB  = HX  + N;       // r / scratch real field
    float* T1R = RB  + N;       // pass1 -> pass2 complex ping
    float* T1I = T1R + N;
    float* T2R = T1I + N;       // pass2 -> pass3 complex (X path)
    float* T2I = T2R + N;
    float* T3R = T2I + N;       // pass2 -> pass3 complex (H(X) path)
    float* T3I = T3R + N;
    float* HFR = T3I + N;       // Hf spectrum, line layout [k_c][k_r]
    float* HFI = HFR + M;
    float* RDV = HFI + M;       // Rdiv, line layout
    float* HPD = RDV + M;       // padded h

    const int GN  = NPLANE / 256;           // element-wise grid
    const int GF  = BATCH * DIM;            // FFT-pipeline grid (8192 lines)
    const float IS = 1.0f / 1024.0f;        // per-pass inverse scale (1/N each dim)

    // State init (every call -> deterministic)
    (void)hipMemsetAsync(X,  0, N * 4, stream);
    (void)hipMemsetAsync(A1, 0, N * 4, stream);
    (void)hipMemsetAsync(A2, 0, 2 * N * 4, stream);
    (void)hipMemsetAsync(A3, 0, N * 4, stream);
    (void)hipMemsetAsync(HX, 0, N * 4, stream);
    (void)hipMemsetAsync(HPD, 0, M * 4, stream);

    // Hf = (-1)^(k0+k1) * FFT2(ct(h))  (parity folds the ifftshift), then Rdiv
    k_embed_h<<<(512 * 512) / 256, 256, 0, stream>>>(h, HPD);
    k_fft_pass1<<<DIM, NT, 0, stream>>>(HPD, T1R, T1I);
    k_fft_pass2<<<DIM, NT, 0, stream>>>(T1R, T1I, HFR, HFI,
                                        nullptr, nullptr, 0,
                                        nullptr, nullptr, nullptr, nullptr,
                                        /*storeSpec*/1, 1.0f);
    k_make_rdiv<<<MPLANE / 256, 256, 0, stream>>>(HFR, HFI, RDV);

    for (int it = 0; it < 20; ++it) {
        k_U<<<GN, 256, 0, stream>>>(X, A2, U);
        k_V<<<GN, 256, 0, stream>>>(A1, HX, y, V);          // HX = H(X_old) cache
        k_W<<<GN, 256, 0, stream>>>(A3, X, W);

        // Hadj(MU1*V - a1) -> RB
        k_q<<<GN, 256, 0, stream>>>(V, A1, RB);
        k_fft_pass1<<<GF, NT, 0, stream>>>(RB, T1R, T1I);
        k_fft_pass2<<<GF, NT, 0, stream>>>(T1R, T1I, T2R, T2I,
                                           HFR, HFI, /*conj*/1,
                                           nullptr, nullptr, nullptr, nullptr,
                                           0, IS);
        k_fft_pass3<<<GF, NT, 0, stream>>>(T2R, T2I, RB, IS);

        // r = (MU3*W - a3) + psit(MU2*U - a2) + Hadj(...)
        k_r<<<GN, 256, 0, stream>>>(U, A2, W, A3, RB);

        // X = real(ifft2(Rdiv*fft2(r))) and H(X) = real(ifft2(Hf*Rdiv*fft2(r)))
        // share one forward FFT2 (dual-filter pass 2).
        k_fft_pass1<<<GF, NT, 0, stream>>>(RB, T1R, T1I);
        k_fft_pass2<<<GF, NT, 0, stream>>>(T1R, T1I, T2R, T2I,
                                           RDV, nullptr, 0,
                                           HFR, HFI, T3R, T3I,
                                           0, IS);
        k_fft_pass3<<<GF, NT, 0, stream>>>(T2R, T2I, X,  IS);
        k_fft_pass3<<<GF, NT, 0, stream>>>(T3R, T3I, HX, IS);

        k_a<<<GN, 256, 0, stream>>>(X, U, V, W, HX, A1, A2, A3);
    }

    k_crop<<<(BATCH * 512 * 512) / 256, 256, 0, stream>>>(X, out);
}